// RoundRobinGate_12515534700961
// MI455X (gfx1250) — compile-verified
//
#include <hip/hip_runtime.h>
#include <stdint.h>

// RoundRobinGate: output = (scalar 0.0, one-hot (4096,16,512) f32, same as bool)
// d_out layout (floats): [0] = 0.0 ; [1 .. 33554432] = onehot ; [33554433 .. 67108864] = bool-as-float
// 268.4 MB of stores, 8192 ones. Pure store-bandwidth problem (~11.5 us at 23.3 TB/s).
// Bulk fill uses the CDNA5 async LDS->global store engine (ASYNCcnt) with NT
// temporal hint: output (268 MB) > L2 (192 MB) and is written exactly once,
// so streaming past L2 is the right cache policy.

#define S_TOK   4096
#define NEXP    16
#define CAP     512                       // 2*S/E
#define ONEHOT  (S_TOK * NEXP * CAP)      // 33,554,432
#define OUTSZ   (1 + 2 * ONEHOT)          // 67,108,865 floats
#define NVEC    (OUTSZ / 4)               // 16,777,216 float4 (covers OUTSZ-1 elements)

#define FILL_BLOCKS   1024
#define FILL_THREADS  256
#define FILL_ITERS    (NVEC / (FILL_BLOCKS * FILL_THREADS))   // 64

// ---------------------------------------------------------------------------
// Kernel A: zero-fill via async LDS->global B128 stores, GVS addressing
// (SGPR64 base + VGPR32 byte offset, whole output < 4 GiB).
// ---------------------------------------------------------------------------
__global__ __launch_bounds__(FILL_THREADS)
void RoundRobinGate_fill_zero(float* __restrict__ out) {
    __shared__ float4 zbuf[FILL_THREADS];
    const int tid = threadIdx.x;
    zbuf[tid] = float4{0.f, 0.f, 0.f, 0.f};
    __syncthreads();

    // Wave-relative LDS byte offset: low 32 bits of the generic pointer.
    uint32_t lds_off = (uint32_t)(uintptr_t)(&zbuf[tid]);

    const uint32_t gid    = blockIdx.x * FILL_THREADS + tid;
    const uint32_t step   = (uint32_t)FILL_BLOCKS * FILL_THREADS * 16u;  // 4 MiB
    uint32_t off32 = gid * 16u;                     // per-lane byte offset
    const uint64_t base = (uint64_t)(uintptr_t)out; // uniform -> SGPR pair

    #pragma unroll 4
    for (int k = 0; k < FILL_ITERS; ++k) {
        // GLOBAL_STORE_ASYNC_FROM_LDS_B128 (GVS): vaddr=off32, vsrc=LDS addr, saddr=base
        asm volatile("global_store_async_from_lds_b128 %0, %1, %2 th:TH_STORE_NT"
                     :
                     : "v"(off32), "v"(lds_off), "s"(base)
                     : "memory");
        off32 += step;
    }
    // S_ENDPGM does an implicit wait-idle, but be explicit about ASYNCcnt.
    asm volatile("s_wait_asynccnt 0" ::: "memory");
}

// ---------------------------------------------------------------------------
// Kernel B: scatter the 8192 ones (+ tail element) after the fill completes.
// Runs after Kernel A on the same stream -> no write races.
// ---------------------------------------------------------------------------
__global__ __launch_bounds__(256)
void RoundRobinGate_set_ones(float* __restrict__ out) {
    const int i = blockIdx.x * blockDim.x + threadIdx.x;   // 0 .. 4095
    if (i >= S_TOK) return;

    if (i == 0) {
        // Final element (index OUTSZ-1) is not covered by the float4 fill.
        __builtin_nontemporal_store(0.0f, &out[OUTSZ - 1]); // [4095,15,511] is not a one
    }

    // output[i, i%E, i//E] = 1.0
    const uint64_t off = 1ull
                       + (uint64_t)i * (NEXP * CAP)
                       + (uint64_t)(i % NEXP) * CAP
                       + (uint64_t)(i / NEXP);
    __builtin_nontemporal_store(1.0f, &out[off]);            // float one-hot
    __builtin_nontemporal_store(1.0f, &out[off + ONEHOT]);   // bool one-hot (as float)
}

extern "C" void kernel_launch(void* const* d_in, const int* in_sizes, int n_in,
                              void* d_out, int out_size, void* d_ws, size_t ws_size,
                              hipStream_t stream) {
    (void)d_in; (void)in_sizes; (void)n_in; (void)d_ws; (void)ws_size; (void)out_size;
    float* out = (float*)d_out;

    RoundRobinGate_fill_zero<<<FILL_BLOCKS, FILL_THREADS, 0, stream>>>(out);
    RoundRobinGate_set_ones<<<(S_TOK + 255) / 256, 256, 0, stream>>>(out);
}